// SimplexPolicy_66795331387580
// MI455X (gfx1250) — compile-verified
//
#include <hip/hip_runtime.h>
#include <math.h>

typedef __attribute__((ext_vector_type(2))) float v2f;
typedef __attribute__((ext_vector_type(8))) float v8f;

#define WAVES_PER_BLOCK 8
#define BLOCK_THREADS   256
#define BATCH_PER_WAVE  8

// padded LDS row strides (floats).
// VH_STRIDE=36: b64 stores at 36*row + 2*n16 -> halves tile banks disjointly
// (8*36 mod 64 = 32); b64 loads at 36*n16 + 2*hi cover all 64 banks once.
// AT_STRIDE=18: 18*m mod 64 distinct for m in 0..15.
#define VH_STRIDE 36
#define AT_STRIDE 18

// K = sqrt(SCALE * log2(e)), SCALE = 1/sqrt(32): folded into gelu output so
// the content WMMA yields log2e-scaled logits and softmax exp is a bare exp2.
#define KFOLD 0.5050426437f

__device__ __forceinline__ v8f wmma4(v2f a, v2f b, v8f c) {
  // D = A(16x4 f32) x B(4x16 f32) + C(16x16 f32)
  return __builtin_amdgcn_wmma_f32_16x16x4_f32(
      /*neg_a=*/false, a, /*neg_b=*/false, b,
      /*c_mod=*/(short)0, c, /*reuse_a=*/false, /*reuse_b=*/false);
}

// Exact-form GELU (branch-free A&S 7.1.26 erf, |err| < 1.5e-7), with the
// output pre-scaled by KFOLD (free: replaces the 0.5 constant).
__device__ __forceinline__ float gelu_scaled(float x) {
  const float z  = x * 0.70710678118654752440f;  // x / sqrt(2)
  const float az = fabsf(z);
  const float t  = __builtin_amdgcn_rcpf(fmaf(0.3275911f, az, 1.0f));
  float p = fmaf(1.061405429f, t, -1.453152027f);
  p = fmaf(p, t, 1.421413741f);
  p = fmaf(p, t, -0.284496736f);
  p = fmaf(p, t, 0.254829592f);
  p *= t;
  const float e = __expf(-az * az);
  const float erf_abs = fmaf(-p, e, 1.0f);
  const float erfv = copysignf(erf_abs, z);
  return (0.5f * KFOLD) * x * (1.0f + erfv);
}

__global__ __launch_bounds__(BLOCK_THREADS)
void simplex_policy_kernel(const float* __restrict__ V,     // (B,16,16)
                           const float* __restrict__ E,     // (B,16,16)
                           const float* __restrict__ SF,    // (B,4)
                           const float* __restrict__ W_vp,  // (16,32)
                           const float* __restrict__ b_vp,  // (32,)
                           const float* __restrict__ alpha_p,
                           const float* __restrict__ W_lh,  // (32,)
                           const float* __restrict__ b_lh_p,
                           const float* __restrict__ W_sb,  // (4,)
                           float* __restrict__ out,         // (B,16)
                           int Btot) {
  __shared__ float lds_vh[WAVES_PER_BLOCK][16 * VH_STRIDE];
  __shared__ float lds_at[WAVES_PER_BLOCK][16 * AT_STRIDE];
  __shared__ float lds_q [WAVES_PER_BLOCK][16];

  const int lane = threadIdx.x & 31;
  const int wave = threadIdx.x >> 5;
  const int n16  = lane & 15;   // column / row-within-tile index
  const int hi   = lane >> 4;   // half-wave select

  float* vh  = lds_vh[wave];
  float* at  = lds_at[wave];
  float* qls = lds_q[wave];

  const float alpha = alpha_p[0];
  const float b_lh  = b_lh_p[0];
  const float elog  = alpha * 1.4426950408889634f;  // alpha * log2(e)

  // ---- per-wave constants (amortized over BATCH_PER_WAVE batches) ----
  // W_vp B-tiles: Bw[t][c] holds W_vp[4c + j + 2*hi][16t + n16], j = component
  v2f Bw[2][4];
#pragma unroll
  for (int t = 0; t < 2; ++t)
#pragma unroll
    for (int c = 0; c < 4; ++c) {
      const int k0 = 4 * c + 2 * hi;
      Bw[t][c].x = W_vp[(k0 + 0) * 32 + 16 * t + n16];
      Bw[t][c].y = W_vp[(k0 + 1) * 32 + 16 * t + n16];
    }
  // W_lh as an all-columns-broadcast B matrix in the *permuted* feature order
  // (p=2j -> col j, p=2j+1 -> col j+16, matching the interleaved vh store),
  // scaled by 1/KFOLD to compensate the KFOLD baked into vh.
  v2f Blh[8];
#pragma unroll
  for (int c = 0; c < 8; ++c) {
    Blh[c].x = W_lh[2 * c + hi]      * (1.0f / KFOLD);
    Blh[c].y = W_lh[2 * c + hi + 16] * (1.0f / KFOLD);
  }
  const float bv0 = b_vp[n16];
  const float bv1 = b_vp[16 + n16];
  const float wsb0 = W_sb[0], wsb1 = W_sb[1], wsb2 = W_sb[2], wsb3 = W_sb[3];
  const v2f ones = {1.0f, 1.0f};  // constant B-tile for row-sum WMMA

  const int wave_id = blockIdx.x * WAVES_PER_BLOCK + wave;
  const int nwaves  = gridDim.x * WAVES_PER_BLOCK;

  for (int b0 = wave_id; b0 < Btot; b0 += nwaves) {
    // b is wave-uniform: scalarize so global addressing is SGPR-base +
    // loop-invariant lane offset (SADDR form), bumping bases on the SALU.
    const int b = __builtin_amdgcn_readfirstlane(b0);
    const float* __restrict__ Vb = V + (size_t)b * 256;
    const float* __restrict__ Eb = E + (size_t)b * 256;

    // ===== 1) vertex_h' = KFOLD * gelu(V @ W_vp + b_vp)  (16x32) =====
    v2f Av[4];
#pragma unroll
    for (int c = 0; c < 4; ++c)
      Av[c] = *(const v2f*)(Vb + n16 * 16 + 4 * c + 2 * hi);

    v8f H0, H1;  // D-layout: element r -> row (r + 8*hi), col (16t + n16)
#pragma unroll
    for (int r = 0; r < 8; ++r) { H0[r] = bv0; H1[r] = bv1; }
#pragma unroll
    for (int c = 0; c < 4; ++c) {
      H0 = wmma4(Av[c], Bw[0][c], H0);
      H1 = wmma4(Av[c], Bw[1][c], H1);
    }
#pragma unroll
    for (int r = 0; r < 8; ++r) {
      H0[r] = gelu_scaled(H0[r]);
      H1[r] = gelu_scaled(H1[r]);
    }

    // stage vh' to LDS, feature-interleaved: one b64 per row instead of
    // two b32 (feature order is irrelevant: every consumer sums over it).
#pragma unroll
    for (int r = 0; r < 8; ++r) {
      v2f pr2 = {H0[r], H1[r]};
      *(v2f*)(vh + (r + 8 * hi) * VH_STRIDE + 2 * n16) = pr2;
    }

    // ===== 2) attn logits (log2-domain): Cc = log2e*(SCALE*vh@vh^T + a*E) ==
    // A-tile c and B-tile c (of vh'^T) have identical per-lane contents.
    v2f Hk[8];
#pragma unroll
    for (int c = 0; c < 8; ++c)
      Hk[c] = *(const v2f*)(vh + n16 * VH_STRIDE + 4 * c + 2 * hi);

    v8f Cc;
#pragma unroll
    for (int r = 0; r < 8; ++r)
      Cc[r] = elog * Eb[(r + 8 * hi) * 16 + n16];
#pragma unroll
    for (int c = 0; c < 8; ++c)
      Cc = wmma4(Hk[c], Hk[c], Cc);

    // ===== 3) q = vh @ W_lh via broadcast-B WMMA (row-broadcast result) ====
    v8f Q = {};
#pragma unroll
    for (int c = 0; c < 8; ++c)
      Q = wmma4(Hk[c], Blh[c], Q);

    // ===== 4) unnormalized softmax numerator: e = exp2(Cc) =====
    // No max-shift: logits are O(5) for this model (unit-normal inputs,
    // 1/sqrt(32) scale), far from exp2 overflow; softmax is shift-invariant.
#pragma unroll
    for (int r = 0; r < 8; ++r)
      Cc[r] = __builtin_exp2f(Cc[r]);

    // stage unnormalized e to LDS; stage q (16 values) from the two
    // column-0 lanes (row-broadcast makes any column valid).
#pragma unroll
    for (int r = 0; r < 8; ++r)
      at[(r + 8 * hi) * AT_STRIDE + n16] = Cc[r];
    if (n16 == 0) {
      float4 q0 = {Q[0], Q[1], Q[2], Q[3]};
      float4 q1 = {Q[4], Q[5], Q[6], Q[7]};
      *(float4*)(qls + 8 * hi)     = q0;
      *(float4*)(qls + 8 * hi + 4) = q1;
    }

    // ===== 5) rowsum S = e @ ones ; t = e @ q (both row-broadcast) =====
    v2f Aa[4];
#pragma unroll
    for (int c = 0; c < 4; ++c)
      Aa[c] = *(const v2f*)(at + n16 * AT_STRIDE + 4 * c + 2 * hi);
    v2f Bq[4];
#pragma unroll
    for (int c = 0; c < 4; ++c)   // broadcast read: same addr per half-wave
      Bq[c] = *(const v2f*)(qls + 4 * c + 2 * hi);

    v8f S = {}, T = {};
#pragma unroll
    for (int c = 0; c < 4; ++c) {
      S = wmma4(Aa[c], ones,  S);
      T = wmma4(Aa[c], Bq[c], T);
    }

    // ===== 6) logits[m] = t[m]/rowsum[m] + q[m] + b_lh + SF@W_sb =====
    const float4 sfv = *(const float4*)(SF + (size_t)b * 4);
    const float sbias = b_lh + sfv.x * wsb0 + sfv.y * wsb1 +
                               sfv.z * wsb2 + sfv.w * wsb3;
    float L[8];
#pragma unroll
    for (int r = 0; r < 8; ++r) {
      const float rinv = __builtin_amdgcn_rcpf(S[r]);
      L[r] = fmaf(T[r], rinv, Q[r] + sbias);
    }
    // all lanes hold the full row-broadcast result; two predicated b128
    // stores from lanes 0 and 16 cover all 16 outputs.
    if (n16 == 0) {
      float* dst = out + (size_t)b * 16 + 8 * hi;
      float4 o0 = {L[0], L[1], L[2], L[3]};
      float4 o1 = {L[4], L[5], L[6], L[7]};
      *(float4*)(dst)     = o0;
      *(float4*)(dst + 4) = o1;
    }
  }
}

extern "C" void kernel_launch(void* const* d_in, const int* in_sizes, int n_in,
                              void* d_out, int out_size, void* d_ws, size_t ws_size,
                              hipStream_t stream) {
  const float* V     = (const float*)d_in[0];
  const float* E     = (const float*)d_in[1];
  const float* SF    = (const float*)d_in[2];
  const float* W_vp  = (const float*)d_in[3];
  const float* b_vp  = (const float*)d_in[4];
  const float* alpha = (const float*)d_in[5];
  const float* W_lh  = (const float*)d_in[6];
  const float* b_lh  = (const float*)d_in[7];
  const float* W_sb  = (const float*)d_in[8];
  float* out = (float*)d_out;

  const int Btot = in_sizes[0] / 256;  // V is (B,16,16)
  const int per_block = WAVES_PER_BLOCK * BATCH_PER_WAVE;
  int blocks = (Btot + per_block - 1) / per_block;
  if (blocks < 1) blocks = 1;

  simplex_policy_kernel<<<blocks, BLOCK_THREADS, 0, stream>>>(
      V, E, SF, W_vp, b_vp, alpha, W_lh, b_lh, W_sb, out, Btot);
}